// STGSN_Dec_63737314672990
// MI455X (gfx1250) — compile-verified
//
#include <hip/hip_runtime.h>

// ---------------------------------------------------------------------------
// Problem constants (compile-time; reference fixes N=16384, EMB_DIM=128)
// ---------------------------------------------------------------------------
#define NN   16384      // nodes
#define D2   256        // 2*EMB_DIM (feature dim of dyn_emb)

typedef __attribute__((ext_vector_type(2))) float v2f;
typedef __attribute__((ext_vector_type(4))) float f4;
typedef __attribute__((ext_vector_type(8))) float v8f;

static __device__ __forceinline__ float fast_exp2(float x) {
#if __has_builtin(__builtin_amdgcn_exp2f)
  return __builtin_amdgcn_exp2f(x);        // v_exp_f32
#else
  return exp2f(x);
#endif
}
static __device__ __forceinline__ float fast_rcp(float x) {
#if __has_builtin(__builtin_amdgcn_rcpf)
  return __builtin_amdgcn_rcpf(x);         // v_rcp_f32
#else
  return 1.0f / x;
#endif
}

// ---------------------------------------------------------------------------
// Phase 1: col = E@W1, rowb = E@W2 + b  via V_WMMA_F32_16X16X4_F32.
// One wave per 16-row tile of E (1024 tiles). B-matrix column 0 carries W1,
// column 1 carries W2, all other columns zero, so after accumulating over
// K=256 the D-matrix column 0 is col[tile] and column 1 is row[tile].
//
// VGPR layouts (ISA 7.12.2, wave32):
//   A 16x4 f32:  lanes 0-15 -> M=lane,    vgpr0/1 = K=0,1
//                lanes 16-31-> M=lane-16, vgpr0/1 = K=2,3
//   B 4x16 f32:  lanes 0-15 -> N=lane,    vgpr0/1 = K=0,1
//                lanes 16-31-> N=lane-16, vgpr0/1 = K=2,3
//   C/D 16x16 f32: lane n (0-15) vgpr r = D[r][n]; lane n+16 vgpr r = D[r+8][n]
// ---------------------------------------------------------------------------
__global__ __launch_bounds__(256) void gemv_wmma(
    const float* __restrict__ emb,   // [NN, D2]
    const float* __restrict__ W,     // [2*D2, 1]
    const float* __restrict__ bias,  // [1]
    float* __restrict__ colv,        // [NN]  (ws)
    float* __restrict__ rowb)        // [NN]  (ws) row + b
{
  const int lane = threadIdx.x & 31;
  const int wave = threadIdx.x >> 5;
  const int tile = blockIdx.x * 8 + wave;   // 0..1023
  const int m0   = tile * 16;

  const int n    = lane & 15;               // matrix row (A) / column (B,D)
  const int koff = (lane >> 4) * 2;         // 0 for lanes 0-15, 2 for 16-31
  const long long arow = (long long)(m0 + n) * D2;

  const float* W1 = W;
  const float* W2 = W + D2;

  v8f acc = {};
  #pragma unroll 4
  for (int kb = 0; kb < D2; kb += 4) {
    const int k = kb + koff;                // even -> 8B aligned v2f loads
    v2f a  = *(const v2f*)(emb + arow + k);
    v2f w1 = *(const v2f*)(W1 + k);
    v2f w2 = *(const v2f*)(W2 + k);
    v2f bb;
    bb.x = (n == 0) ? w1.x : ((n == 1) ? w2.x : 0.0f);
    bb.y = (n == 0) ? w1.y : ((n == 1) ? w2.y : 0.0f);
    // (neg_a, A, neg_b, B, c_mod, C, reuse_a, reuse_b)
    acc = __builtin_amdgcn_wmma_f32_16x16x4_f32(
        false, a, false, bb, (short)0, acc, false, false);
  }

  // D column 0 = col values, column 1 = row values.
  const float bval = bias[0];
  if (n <= 1) {
    float* dst = (n == 0) ? colv : rowb;
    const float add = (n == 0) ? 0.0f : bval;
    const int mm = m0 + (lane >> 4) * 8;    // lanes 0/1 -> M=0..7, 16/17 -> 8..15
    #pragma unroll
    for (int r = 0; r < 8; ++r) dst[mm + r] = acc[r] + add;
  }
}

// ---------------------------------------------------------------------------
// Phase 2: out[i][j] = sigmoid(rowb[i] + colv[j]).  Store-bound: 1.07 GB fp32
// streamed once -> non-temporal 128-bit stores, wave-contiguous 512B bursts.
// sigmoid(x) = rcp(1 + exp2(-x * log2(e))): 1 trans exp + 1 trans rcp / elem.
// Grid: x = row (16384), y = column chunk (4 x 4096 floats).
// Each thread: 4 float4 at stride 256*16B so each store instruction across a
// wave covers 512 consecutive bytes.
// ---------------------------------------------------------------------------
__global__ __launch_bounds__(256) void outer_sigmoid(
    const float* __restrict__ rowb,
    const f4*    __restrict__ col4,   // colv viewed as float4 [NN/4]
    f4*          __restrict__ out4)   // output viewed as float4 [NN * NN/4]
{
  const int  i     = blockIdx.x;                       // row
  const int  base4 = blockIdx.y * 1024 + threadIdx.x;  // float4 index in row
  const long long rowbase = (long long)i * (NN / 4);
  const float rv = rowb[i];                            // uniform per block

  constexpr float kLog2e = 1.4426950408889634f;

  #pragma unroll
  for (int kk = 0; kk < 4; ++kk) {
    const int j4 = base4 + kk * 256;
    f4 c = col4[j4];                                   // L2-resident (64 KB)
    f4 s;
    #pragma unroll
    for (int e = 0; e < 4; ++e) {
      const float x  = rv + c[e];
      const float ex = fast_exp2(x * -kLog2e);         // e^{-x}
      s[e] = fast_rcp(1.0f + ex);                      // 1/(1+e^{-x})
    }
    __builtin_nontemporal_store(s, out4 + rowbase + j4);
  }
}

// ---------------------------------------------------------------------------
extern "C" void kernel_launch(void* const* d_in, const int* in_sizes, int n_in,
                              void* d_out, int out_size, void* d_ws, size_t ws_size,
                              hipStream_t stream) {
  const float* emb  = (const float*)d_in[0];   // [16384, 256] f32
  const float* W    = (const float*)d_in[1];   // [512, 1]     f32
  const float* bias = (const float*)d_in[2];   // [1]          f32
  // d_in[3] = num_nodes (int scalar) -- compile-time NN here.

  float* colv = (float*)d_ws;                  // 16384 f32
  float* rowb = colv + NN;                     // 16384 f32  (128 KB total)

  // Phase 1: 1024 tiles, 8 waves/block -> 128 blocks.
  gemv_wmma<<<128, 256, 0, stream>>>(emb, W, bias, colv, rowb);

  // Phase 2: 16384 rows x 4 column-chunks of 4096 floats.
  outer_sigmoid<<<dim3(NN, 4), 256, 0, stream>>>(
      rowb, (const f4*)colv, (f4*)d_out);
}